// SlotAttention_32083405701250
// MI455X (gfx1250) — compile-verified
//
#include <hip/hip_runtime.h>
#include <math.h>
#include <stdint.h>

typedef __attribute__((ext_vector_type(2))) float v2f;
typedef __attribute__((ext_vector_type(4))) float v4f;
typedef __attribute__((ext_vector_type(8))) float v8f;

#define DEV __device__ __forceinline__

constexpr int cB = 64, cN = 4096, cDIN = 256, cD = 256, cK = 8, cH = 512, NITER = 3;
constexpr float EPS_LN = 1e-5f, EPS_ATTN = 1e-8f, SCALE = 0.0625f; // 1/sqrt(256)

// ---------------- workspace layout (offsets in floats) ----------------
constexpr size_t off_k     = 0;                                   // [B*N, 256] row-major f32
constexpr size_t off_v     = off_k + (size_t)cB * cN * cD;        // [B*N/2, 256] of float2 (n-pair packed)
constexpr size_t off_q     = off_v + (size_t)cB * cN * cD;        // [B, 16, 256] (rows 8..15 zero)
constexpr size_t off_slots = off_q + (size_t)cB * 16 * cD;        // [B, 8, 256]
constexpr size_t off_sums  = off_slots + (size_t)cB * cK * cD;    // [B, 8]
constexpr size_t off_upd   = off_sums + (size_t)cB * cK;          // [B, 8, 256]
constexpr size_t off_part  = off_upd + (size_t)cB * cK * cD;      // [32, B, 8, 256]
constexpr size_t off_BPk   = off_part + 32ull * cB * cK * cD;     // packed Wk fragments
constexpr size_t off_BPv   = off_BPk + 64 * 16 * 32 * 2;          // packed Wv fragments
constexpr size_t off_WqT   = off_BPv + 64 * 16 * 32 * 2;          // [256,256]
constexpr size_t off_WihT  = off_WqT + 256 * 256;                 // [256,768]
constexpr size_t off_WhhT  = off_WihT + 256 * 768;                // [256,768]
constexpr size_t off_W1T   = off_WhhT + 256 * 768;                // [256,512]
constexpr size_t off_W2T   = off_W1T + 256 * 512;                 // [512,256]

// ---------------- CDNA5 f32 WMMA: D = A(16x4) * B(4x16) + C ----------------
DEV v8f wmma4(v2f a, v2f b, v8f c) {
  return __builtin_amdgcn_wmma_f32_16x16x4_f32(false, a, false, b, (short)0, c,
                                               false, false);
}

DEV float wave_sum(float x) {
#pragma unroll
  for (int o = 16; o > 0; o >>= 1) x += __shfl_xor(x, o, 32);
  return x;
}

// CDNA5 async DMA: global -> LDS, 16B per lane, tracked by ASYNCcnt.
// INST_OFFSET is added to BOTH the LDS and global addresses (ISA 08 §4.4).
DEV void async_g2l_b128(uint32_t lds_byte_addr, const void* gaddr) {
  asm volatile("global_load_async_to_lds_b128 %0, %1, off"
               :: "v"(lds_byte_addr), "v"((uint64_t)(uintptr_t)gaddr)
               : "memory");
}
DEV void async_g2l_b128_off16(uint32_t lds_byte_addr, const void* gaddr) {
  asm volatile("global_load_async_to_lds_b128 %0, %1, off offset:16"
               :: "v"(lds_byte_addr), "v"((uint64_t)(uintptr_t)gaddr)
               : "memory");
}
DEV void wait_async0() { asm volatile("s_wait_asynccnt 0x0" ::: "memory"); }

// ---------------- weight prep ----------------
__global__ void pack_frag_kernel(const float* __restrict__ W, float* __restrict__ BP) {
  int idx  = blockIdx.x * 256 + threadIdx.x;     // 0 .. 64*16*32-1
  int lane = idx & 31;
  int nt   = (idx >> 5) & 15;
  int kk4  = idx >> 9;
  int j    = nt * 16 + (lane & 15);              // output column (W row)
  int c    = kk4 * 4 + ((lane < 16) ? 0 : 2);    // K index
  BP[idx * 2 + 0] = W[j * 256 + c];
  BP[idx * 2 + 1] = W[j * 256 + c + 1];
}

__global__ void transpose_kernel(const float* __restrict__ W, float* __restrict__ WT,
                                 int rows, int cols) {
  int idx = blockIdx.x * 256 + threadIdx.x;
  if (idx >= rows * cols) return;
  int r = idx / cols, c = idx - r * cols;
  WT[c * rows + r] = W[idx];
}

__global__ void init_slots_kernel(const float* __restrict__ noise,
                                  const float* __restrict__ mu,
                                  const float* __restrict__ ls,
                                  float* __restrict__ slots) {
  int idx = blockIdx.x * 256 + threadIdx.x;      // B*K*D = 131072
  int d   = idx & 255;
  slots[idx] = mu[d] + expf(ls[d]) * noise[idx];
}

// ---------------- LN(inputs) + k/v projection (the big WMMA GEMM) ----------------
// grid 4096 blocks x 256 thr; block tile: 64 rows x 256 cols for BOTH k and v.
__global__ void __launch_bounds__(256, 1)
lnkv_kernel(const float* __restrict__ inp, const float* __restrict__ g,
            const float* __restrict__ bln, const float* __restrict__ bk,
            const float* __restrict__ bv, float* __restrict__ ws) {
  extern __shared__ float xs[];                  // [64][260] padded, bank-conflict free
  const int t = threadIdx.x, w = t >> 5, lane = t & 31;
  const int blk = blockIdx.x;

  // Phase 1: layernorm 64 input rows into LDS (wave handles 8 rows)
  for (int i = 0; i < 8; ++i) {
    int row = w * 8 + i;
    const float* rp = inp + (size_t)(blk * 64 + row) * 256 + lane * 8;
    v4f x0 = *(const v4f*)rp;
    v4f x1 = *(const v4f*)(rp + 4);
    float x[8] = {x0[0], x0[1], x0[2], x0[3], x1[0], x1[1], x1[2], x1[3]};
    float s = 0.f;
#pragma unroll
    for (int j = 0; j < 8; ++j) s += x[j];
    float mean = wave_sum(s) * (1.0f / 256.0f);
    float s2 = 0.f;
#pragma unroll
    for (int j = 0; j < 8; ++j) { float d = x[j] - mean; s2 += d * d; }
    float rstd = 1.0f / sqrtf(wave_sum(s2) * (1.0f / 256.0f) + EPS_LN);
#pragma unroll
    for (int j = 0; j < 8; ++j) {
      int c = lane * 8 + j;
      xs[row * 260 + c] = (x[j] - mean) * rstd * g[c] + bln[c];
    }
  }
  __syncthreads();

  // Phase 2: wave w -> {k|v} x 16-row M-subtile, full 256 cols
  const int mat = w >> 2, msub = w & 3;
  const int koff = (lane < 16) ? 0 : 2;
  const float* ap = &xs[(msub * 16 + (lane & 15)) * 260 + koff];
  const v2f* bp = (const v2f*)(ws + (mat ? off_BPv : off_BPk)) + lane;
  v8f acc[16];
#pragma unroll
  for (int nt = 0; nt < 16; ++nt) acc[nt] = (v8f){0.f,0.f,0.f,0.f,0.f,0.f,0.f,0.f};

  for (int kk4 = 0; kk4 < 64; ++kk4) {
    v2f a = *(const v2f*)(ap + kk4 * 4);
    const v2f* bq = bp + (size_t)kk4 * 16 * 32;
#pragma unroll
    for (int nt = 0; nt < 16; ++nt) acc[nt] = wmma4(a, bq[nt * 32], acc[nt]);
  }

  // Phase 3: add bias, store. C layout: VGPR r, lanes<16 -> M=r, lanes>=16 -> M=r+8.
  const int rowbase = blk * 64 + msub * 16 + ((lane >= 16) ? 8 : 0);
  const int col0 = lane & 15;
  if (mat == 0) {                                 // k: row-major scalar stores
    float* out = ws + off_k;
#pragma unroll
    for (int nt = 0; nt < 16; ++nt) {
      int col = nt * 16 + col0;
      float bb = bk[col];
#pragma unroll
      for (int r = 0; r < 8; ++r)
        out[(size_t)(rowbase + r) * 256 + col] = acc[nt][r] + bb;
    }
  } else {                                        // v: n-pair packed float2, coalesced b64
    v2f* vP = (v2f*)(ws + off_v);
#pragma unroll
    for (int nt = 0; nt < 16; ++nt) {
      int col = nt * 16 + col0;
      float bb = bv[col];
#pragma unroll
      for (int i = 0; i < 4; ++i) {
        v2f pr; pr.x = acc[nt][2 * i] + bb; pr.y = acc[nt][2 * i + 1] + bb;
        vP[(size_t)(rowbase / 2 + i) * 256 + col] = pr;
      }
    }
  }
}

// ---------------- per-iteration: LN(slots) + q projection ----------------
__global__ void __launch_bounds__(256, 1)
slots_q_kernel(float* __restrict__ ws, const float* __restrict__ gs,
               const float* __restrict__ bs, const float* __restrict__ bq) {
  __shared__ float sn[8][256];
  const int b = blockIdx.x, t = threadIdx.x, w = t >> 5, lane = t & 31;
  const float* slots = ws + off_slots + (size_t)b * cK * cD;
  {
    const float* rp = slots + w * 256 + lane * 8;
    float x[8];
#pragma unroll
    for (int j = 0; j < 8; ++j) x[j] = rp[j];
    float s = 0.f;
#pragma unroll
    for (int j = 0; j < 8; ++j) s += x[j];
    float mean = wave_sum(s) * (1.0f / 256.0f);
    float s2 = 0.f;
#pragma unroll
    for (int j = 0; j < 8; ++j) { float d = x[j] - mean; s2 += d * d; }
    float rstd = 1.0f / sqrtf(wave_sum(s2) * (1.0f / 256.0f) + EPS_LN);
#pragma unroll
    for (int j = 0; j < 8; ++j) {
      int c = lane * 8 + j;
      sn[w][c] = (x[j] - mean) * rstd * gs[c] + bs[c];
    }
  }
  __syncthreads();
  const float* WqT = ws + off_WqT;
  float* q = ws + off_q + (size_t)b * 16 * 256;
  float a[8];
#pragma unroll
  for (int row = 0; row < 8; ++row) a[row] = 0.f;
  for (int c = 0; c < 256; ++c) {
    float wq = WqT[c * 256 + t];                  // 1 coalesced load -> 8 FMA
#pragma unroll
    for (int row = 0; row < 8; ++row) a[row] += sn[row][c] * wq;
  }
#pragma unroll
  for (int row = 0; row < 8; ++row) q[row * 256 + t] = a[row] + bq[t];
#pragma unroll
  for (int row = 8; row < 16; ++row) q[row * 256 + t] = 0.0f;  // zero-pad M
}

// ---------------- dots = q k^T, softmax over slots, write attn ----------------
// grid (64 b x 32 chunks) x 256 thr; wave -> 16 n's; slots for a column live in one lane.
// k tile staged to LDS with CDNA5 async direct-to-LDS DMA loads.
__global__ void __launch_bounds__(256, 1)
dots_kernel(const float* __restrict__ ws, float* __restrict__ attn) {
  extern __shared__ float ksh[];                 // [128][260]
  const int t = threadIdx.x, w = t >> 5, lane = t & 31;
  const int b = blockIdx.x >> 5, chunk = blockIdx.x & 31;
  const int n0 = chunk * 128;
  const float* kbase = ws + off_k + ((size_t)b * cN + n0) * 256;
  {
    const uint32_t lbase = (uint32_t)(uintptr_t)(&ksh[0]);  // LDS aperture: addr[31:0]
#pragma unroll
    for (int i = 0; i < 16; ++i) {
      int row = w * 16 + i;
      const float* ga = kbase + (size_t)row * 256 + lane * 8;
      uint32_t la = lbase + (uint32_t)(row * 260 + lane * 8) * 4u;
      async_g2l_b128(la, ga);                    // bytes [0,16)  of this lane's 32B
      async_g2l_b128_off16(la, ga);              // bytes [16,32)
    }
    wait_async0();
  }
  __syncthreads();

  const int koff = (lane < 16) ? 0 : 2;
  const float* ap = ws + off_q + ((size_t)b * 16 + (lane & 15)) * 256 + koff;
  const float* bp = &ksh[(w * 16 + (lane & 15)) * 260 + koff];
  v8f acc = (v8f){0.f,0.f,0.f,0.f,0.f,0.f,0.f,0.f};
  for (int kk4 = 0; kk4 < 64; ++kk4)
    acc = wmma4(*(const v2f*)(ap + kk4 * 4), *(const v2f*)(bp + kk4 * 4), acc);

  if (lane < 16) {                               // m=0..7 live in acc[0..7] of this lane
    float dv[8];
    float mx = -3.0e38f;
#pragma unroll
    for (int r = 0; r < 8; ++r) { dv[r] = acc[r] * SCALE; mx = fmaxf(mx, dv[r]); }
    float s = 0.f;
#pragma unroll
    for (int r = 0; r < 8; ++r) { dv[r] = expf(dv[r] - mx); s += dv[r]; }
    float inv = 1.0f / s;
    int n = n0 + w * 16 + lane;
#pragma unroll
    for (int r = 0; r < 8; ++r)
      attn[((size_t)b * 8 + r) * 4096 + n] = dv[r] * inv;
  }
}

// ---------------- deterministic per-(b,m) sums of attn over n ----------------
__global__ void attn_sums_kernel(const float* __restrict__ attn, float* __restrict__ ws) {
  __shared__ float red[256];
  const int bm = blockIdx.x, t = threadIdx.x;
  const float* base = attn + (size_t)bm * 4096;
  float s = 0.f;
  for (int i = t; i < 4096; i += 256) s += base[i];
  red[t] = s;
  __syncthreads();
  for (int o = 128; o > 0; o >>= 1) {
    if (t < o) red[t] += red[t + o];
    __syncthreads();
  }
  if (t == 0) ws[off_sums + bm] = red[0];
}

// ---------------- updates = attn_norm @ v : K-split WMMA partials ----------------
// grid 256 x 256 thr; wave owns one 128-n chunk (32 chunks per batch).
__global__ void __launch_bounds__(256, 1)
updates_partial_kernel(const float* __restrict__ attn, float* __restrict__ ws) {
  const int t = threadIdx.x, w = t >> 5, lane = t & 31;
  const int b = blockIdx.x >> 2, c4 = blockIdx.x & 3;
  const int p = c4 * 8 + w, n0 = p * 128;
  const int m = lane & 15;
  const int half = (lane < 16) ? 0 : 1;
  const int koff = half * 2;
  const float rn = (m < 8) ? 1.0f / (ws[off_sums + b * 8 + m] + EPS_ATTN) : 0.0f;
  const int msafe = (m < 8) ? m : 0;             // rows >=8 multiplied by rn=0
  const float* apt = attn + ((size_t)b * 8 + msafe) * 4096 + n0 + koff;
  // v is n-pair packed: vP[(global_n/2)*256 + col] = {v[n],v[n+1]}
  const v2f* vP = (const v2f*)(ws + off_v) + (((size_t)b * cN + n0) / 2) * 256 + (lane & 15);
  v8f acc[16];
#pragma unroll
  for (int dt = 0; dt < 16; ++dt) acc[dt] = (v8f){0.f,0.f,0.f,0.f,0.f,0.f,0.f,0.f};

  for (int kk4 = 0; kk4 < 32; ++kk4) {
    v2f a = *(const v2f*)(apt + kk4 * 4);
    a.x *= rn; a.y *= rn;                        // fold 1/(sum+eps) into A rows
    const v2f* vrow = vP + (size_t)(2 * kk4 + half) * 256;
#pragma unroll
    for (int dt = 0; dt < 16; ++dt)
      acc[dt] = wmma4(a, vrow[dt * 16], acc[dt]);  // coalesced b64 B-fragment
  }
  if (lane < 16) {
    float* part = ws + off_part + ((size_t)p * 64 + b) * 8 * 256;
#pragma unroll
    for (int dt = 0; dt < 16; ++dt)
#pragma unroll
      for (int r = 0; r < 8; ++r)
        part[(size_t)r * 256 + dt * 16 + lane] = acc[dt][r];
  }
}

__global__ void updates_reduce_kernel(float* __restrict__ ws) {
  const int i = blockIdx.x * 256 + threadIdx.x;  // 0..131071
  float s = 0.f;
  for (int p = 0; p < 32; ++p) s += ws[off_part + (size_t)p * 131072 + i];
  ws[off_upd + i] = s;
}

// ---------------- GRU cell + LN + MLP residual (per batch) ----------------
__global__ void __launch_bounds__(256, 1)
gru_mlp_kernel(float* __restrict__ ws, const float* __restrict__ bih,
               const float* __restrict__ bhh, const float* __restrict__ gf,
               const float* __restrict__ bfp, const float* __restrict__ b1,
               const float* __restrict__ b2) {
  __shared__ float up[8][256], sp[8][256], hh[8][256], ffn[8][256], act[8][512];
  const int b = blockIdx.x, t = threadIdx.x, w = t >> 5, lane = t & 31;
  float* slots = ws + off_slots + (size_t)b * 2048;
  const float* upd = ws + off_upd + (size_t)b * 2048;
#pragma unroll
  for (int j = 0; j < 8; ++j) {
    up[w][lane * 8 + j] = upd[w * 256 + lane * 8 + j];
    sp[w][lane * 8 + j] = slots[w * 256 + lane * 8 + j];
  }
  __syncthreads();

  const float* WihT = ws + off_WihT;
  const float* WhhT = ws + off_WhhT;
  {
    float ir[8], iz[8], inn[8], hr[8], hz[8], hn[8];
#pragma unroll
    for (int r = 0; r < 8; ++r) { ir[r]=iz[r]=inn[r]=hr[r]=hz[r]=hn[r]=0.f; }
    for (int c = 0; c < 256; ++c) {
      const float* wi = WihT + c * 768;
      const float* wh = WhhT + c * 768;
      float wi0 = wi[t], wi1 = wi[256 + t], wi2 = wi[512 + t];
      float wh0 = wh[t], wh1 = wh[256 + t], wh2 = wh[512 + t];
#pragma unroll
      for (int r = 0; r < 8; ++r) {
        float ux = up[r][c], hx = sp[r][c];
        ir[r] += ux * wi0; iz[r] += ux * wi1; inn[r] += ux * wi2;
        hr[r] += hx * wh0; hz[r] += hx * wh1; hn[r]  += hx * wh2;
      }
    }
    float bi0 = bih[t], bi1 = bih[256 + t], bi2 = bih[512 + t];
    float bh0 = bhh[t], bh1 = bhh[256 + t], bh2 = bhh[512 + t];
#pragma unroll
    for (int r = 0; r < 8; ++r) {
      float rr = 1.0f / (1.0f + expf(-(ir[r] + bi0 + hr[r] + bh0)));
      float zz = 1.0f / (1.0f + expf(-(iz[r] + bi1 + hz[r] + bh1)));
      float nn = tanhf(inn[r] + bi2 + rr * (hn[r] + bh2));
      hh[r][t] = (1.0f - zz) * nn + zz * sp[r][t];
    }
  }
  __syncthreads();

  {  // layernorm hh row w -> ffn
    float x[8];
#pragma unroll
    for (int j = 0; j < 8; ++j) x[j] = hh[w][lane * 8 + j];
    float s = 0.f;
#pragma unroll
    for (int j = 0; j < 8; ++j) s += x[j];
    float mean = wave_sum(s) * (1.0f / 256.0f);
    float s2 = 0.f;
#pragma unroll
    for (int j = 0; j < 8; ++j) { float d = x[j] - mean; s2 += d * d; }
    float rstd = 1.0f / sqrtf(wave_sum(s2) * (1.0f / 256.0f) + EPS_LN);
#pragma unroll
    for (int j = 0; j < 8; ++j) {
      int c = lane * 8 + j;
      ffn[w][c] = (x[j] - mean) * rstd * gf[c] + bfp[c];
    }
  }
  __syncthreads();

  const float* W1T = ws + off_W1T;
  {
    float a0[8], a1[8];
#pragma unroll
    for (int r = 0; r < 8; ++r) { a0[r] = 0.f; a1[r] = 0.f; }
    for (int c = 0; c < 256; ++c) {
      float w1a = W1T[c * 512 + t], w1b = W1T[c * 512 + 256 + t];
#pragma unroll
      for (int r = 0; r < 8; ++r) {
        float fx = ffn[r][c];
        a0[r] += fx * w1a; a1[r] += fx * w1b;
      }
    }
    float bb0 = b1[t], bb1 = b1[256 + t];
#pragma unroll
    for (int r = 0; r < 8; ++r) {
      act[r][t] = fmaxf(a0[r] + bb0, 0.0f);
      act[r][256 + t] = fmaxf(a1[r] + bb1, 0.0f);
    }
  }
  __syncthreads();

  const float* W2T = ws + off_W2T;
  {
    float o[8];
#pragma unroll
    for (int r = 0; r < 8; ++r) o[r] = 0.f;
    for (int h = 0; h < 512; ++h) {
      float w2 = W2T[h * 256 + t];
#pragma unroll
      for (int r = 0; r < 8; ++r) o[r] += act[r][h] * w2;
    }
    float bb = b2[t];
#pragma unroll
    for (int r = 0; r < 8; ++r) slots[r * 256 + t] = hh[r][t] + o[r] + bb;
  }
}

__global__ void copy_slots_kernel(const float* __restrict__ ws, float* __restrict__ out) {
  int i = blockIdx.x * 256 + threadIdx.x;
  out[i] = ws[off_slots + i];
}

// ---------------- host launcher ----------------
extern "C" void kernel_launch(void* const* d_in, const int* in_sizes, int n_in,
                              void* d_out, int out_size, void* d_ws, size_t ws_size,
                              hipStream_t stream) {
  (void)in_sizes; (void)n_in; (void)out_size; (void)ws_size;
  const float* inputs  = (const float*)d_in[0];
  const float* noise   = (const float*)d_in[1];
  const float* slot_mu = (const float*)d_in[2];
  const float* slot_ls = (const float*)d_in[3];
  const float* Wq  = (const float*)d_in[4];
  const float* bq  = (const float*)d_in[5];
  const float* Wk  = (const float*)d_in[6];
  const float* bk  = (const float*)d_in[7];
  const float* Wv  = (const float*)d_in[8];
  const float* bv  = (const float*)d_in[9];
  const float* Wih = (const float*)d_in[10];
  const float* bih = (const float*)d_in[11];
  const float* Whh = (const float*)d_in[12];
  const float* bhh = (const float*)d_in[13];
  const float* gin = (const float*)d_in[14];
  const float* bin = (const float*)d_in[15];
  const float* gs  = (const float*)d_in[16];
  const float* bs  = (const float*)d_in[17];
  const float* gff = (const float*)d_in[18];
  const float* bff = (const float*)d_in[19];
  const float* W1  = (const float*)d_in[20];
  const float* b1  = (const float*)d_in[21];
  const float* W2  = (const float*)d_in[22];
  const float* b2  = (const float*)d_in[23];

  float* ws = (float*)d_ws;
  float* out_slots = (float*)d_out;
  float* attn = out_slots + (size_t)cB * cK * cD;   // output layout: slots | attn

  pack_frag_kernel<<<128, 256, 0, stream>>>(Wk, ws + off_BPk);
  pack_frag_kernel<<<128, 256, 0, stream>>>(Wv, ws + off_BPv);
  transpose_kernel<<<256, 256, 0, stream>>>(Wq,  ws + off_WqT, 256, 256);
  transpose_kernel<<<768, 256, 0, stream>>>(Wih, ws + off_WihT, 768, 256);
  transpose_kernel<<<768, 256, 0, stream>>>(Whh, ws + off_WhhT, 768, 256);
  transpose_kernel<<<512, 256, 0, stream>>>(W1,  ws + off_W1T, 512, 256);
  transpose_kernel<<<512, 256, 0, stream>>>(W2,  ws + off_W2T, 256, 512);
  init_slots_kernel<<<512, 256, 0, stream>>>(noise, slot_mu, slot_ls, ws + off_slots);

  lnkv_kernel<<<4096, 256, 64 * 260 * 4, stream>>>(inputs, gin, bin, bk, bv, ws);

  for (int it = 0; it < NITER; ++it) {
    slots_q_kernel<<<64, 256, 0, stream>>>(ws, gs, bs, bq);
    dots_kernel<<<2048, 256, 128 * 260 * 4, stream>>>(ws, attn);
    attn_sums_kernel<<<512, 256, 0, stream>>>(attn, ws);
    updates_partial_kernel<<<256, 256, 0, stream>>>(attn, ws);
    updates_reduce_kernel<<<512, 256, 0, stream>>>(ws);
    gru_mlp_kernel<<<64, 256, 0, stream>>>(ws, bih, bhh, gff, bff, b1, b2);
  }
  copy_slots_kernel<<<512, 256, 0, stream>>>(ws, out_slots);
}